// Seg2dTopk_51153060495483
// MI455X (gfx1250) — compile-verified
//
#include <hip/hip_runtime.h>
#include <math.h>

// ---------------------------------------------------------------------------
// MISE adaptive occupancy on gfx1250 (MI455X).
//  * Layer-1 GEMM on V_WMMA_F32_16X16X4_F32, bias folded into B's K=2 row
//    (A rows are (x, y, 1, 0)) -> exact f32 affine layer, 1 WMMA / 16x16 tile.
//  * Layer-2 (dot over 256) on the WMMA C/D register layout: v_med3 ReLU +
//    FMA per tile, then a 4-step wave32 shfl_xor tree per lane-half.
//  * Weights staged per block into LDS via GLOBAL_LOAD_ASYNC_TO_LDS_B128
//    (ASYNCcnt), when the toolchain exposes the builtin.
// ---------------------------------------------------------------------------

typedef __attribute__((ext_vector_type(2))) float v2f;
typedef __attribute__((ext_vector_type(8))) float v8f;
typedef int v4i __attribute__((ext_vector_type(4)));

#define FINAL_RES 1025
#define PLANE (1025 * 1025)
#define HIDDEN 256
#define NBINS 16384

#if defined(__has_builtin)
#if __has_builtin(__builtin_amdgcn_global_load_async_to_lds_b128) && \
    __has_builtin(__builtin_amdgcn_s_wait_asynccnt)
#define HAVE_ASYNC_LDS 1
#endif
#endif
#ifndef HAVE_ASYNC_LDS
#define HAVE_ASYNC_LDS 0
#endif

#if HAVE_ASYNC_LDS
typedef __attribute__((address_space(1))) v4i as1_v4i;  // global int4
typedef __attribute__((address_space(3))) v4i as3_v4i;  // LDS int4
#endif

// LDS layout: [0,512) = w1 (row0 then row1), [512,768) = b1, [768,1024) = w2.
#define SM_W1 0
#define SM_B1 512
#define SM_W2 768

// ---------------------------------------------------------------------------
// MLP evaluation: each wave32 processes 16 points.
//   A (16x4 f32): lanes 0-15 hold (x_m, y_m) for m = lane (K=0,1);
//                 lanes 16-31 hold (1, 0)            (K=2,3).
//   B (4x16 f32): lanes 0-15 hold (w1x[n], w1y[n])   (K=0,1 rows);
//                 lanes 16-31 hold (b1[n], 0)        (K=2,3 rows).
// If idx_list == nullptr, point p's flat index is just p (regular grid).
// ---------------------------------------------------------------------------
__global__ void __launch_bounds__(256) mlp_eval_kernel(
    const float* __restrict__ w1, const float* __restrict__ b1,
    const float* __restrict__ w2, const float* __restrict__ b2,
    const int* __restrict__ idx_list, int n_points, int res, float stride,
    float* __restrict__ occ)
{
    __shared__ float smem[1024];

    const int tid = threadIdx.x;
    const int lane = tid & 31;
    const int wave = blockIdx.x * 8 + (tid >> 5);
    const int base = wave * 16;

    // ---- cooperative weight staging: 1024 floats, 4 per thread (16B) ----
    {
        const float* src = (tid < 128) ? (w1 + tid * 4)
                         : (tid < 192) ? (b1 + (tid - 128) * 4)
                                       : (w2 + (tid - 192) * 4);
#if HAVE_ASYNC_LDS
        // Global generic addr == AS1 addr; generic LDS addr low 32 bits == LDS
        // byte offset (flat-aperture rule), so integer round-trips are exact.
        as1_v4i* gp = (as1_v4i*)(unsigned long long)src;
        as3_v4i* lp = (as3_v4i*)(unsigned int)(unsigned long long)&smem[tid * 4];
        __builtin_amdgcn_global_load_async_to_lds_b128(gp, lp, 0, 0);
        __builtin_amdgcn_s_wait_asynccnt(0);
#else
        *(float4*)&smem[tid * 4] = *(const float4*)src;
#endif
    }
    __syncthreads();

    const int nn = lane & 15;       // neuron-within-tile / point-within-group
    const bool hi = lane >= 16;

    // B operand per 16-col tile, plus w2 gather (per-lane column nn+16t).
    v2f Bt[16];
    float w2t[16];
#pragma unroll
    for (int t = 0; t < 16; ++t) {
        const int col = nn + 16 * t;
        Bt[t].x = hi ? smem[SM_B1 + col] : smem[SM_W1 + col];        // K=0 / K=2
        Bt[t].y = hi ? 0.0f              : smem[SM_W1 + 256 + col];  // K=1 / K=3
        w2t[t] = smem[SM_W2 + col];
    }

    // Point coordinates (clamped tail; stores are guarded, WMMA keeps EXEC=~0).
    int pid = base + nn;
    if (pid > n_points - 1) pid = n_points - 1;
    const int fidx = idx_list ? idx_list[pid] : pid;
    const float px = (float)(fidx % res) * stride;
    const float py = (float)(fidx / res) * stride;
    const float cx = (px + 0.5f) * (2.0f / 1025.0f) - 1.0f;
    const float cy = (py + 0.5f) * (2.0f / 1025.0f) - 1.0f;

    v2f A;
    A.x = hi ? 1.0f : cx;
    A.y = hi ? 0.0f : cy;

    float acc[8];
#pragma unroll
    for (int v = 0; v < 8; ++v) acc[v] = 0.0f;

    const float inf = __builtin_inff();
#pragma unroll
    for (int t = 0; t < 16; ++t) {
        v8f c = {0.f, 0.f, 0.f, 0.f, 0.f, 0.f, 0.f, 0.f};
        v8f d = __builtin_amdgcn_wmma_f32_16x16x4_f32(
            /*neg_a=*/false, A, /*neg_b=*/false, Bt[t],
            /*c_mod=*/(short)0, c, /*reuse_a=*/false, /*reuse_b=*/false);
        // Layer 2 partial: D[v][lane] = h[m, n], m = v + 8*hi, n = nn + 16t.
#pragma unroll
        for (int v = 0; v < 8; ++v) {
            const float h = __builtin_amdgcn_fmed3f(d[v], 0.0f, inf);  // ReLU
            acc[v] += h * w2t[t];
        }
    }

    // Reduce over the 16 columns inside each lane-half (wave32 shuffles).
#pragma unroll
    for (int v = 0; v < 8; ++v) {
        acc[v] += __shfl_xor(acc[v], 8, 32);
        acc[v] += __shfl_xor(acc[v], 4, 32);
        acc[v] += __shfl_xor(acc[v], 2, 32);
        acc[v] += __shfl_xor(acc[v], 1, 32);
    }

    // lane 0 holds rows m = 0..7 (acc[v] = row v); lane 16 holds rows 8..15.
    if (nn == 0) {
        const float bias2 = b2[0];
        const int mbase = base + (hi ? 8 : 0);
#pragma unroll
        for (int v = 0; v < 8; ++v) {
            const int p = mbase + v;
            if (p < n_points) {
                const float logit = acc[v] + bias2;
                const float e = __expf(-logit);
                const float o = __builtin_amdgcn_rcpf(1.0f + e);
                const int f = idx_list ? idx_list[p] : p;
                occ[f] = o;
            }
        }
    }
}

// ---------------------------------------------------------------------------
// Bilinear upsample, align_corners=True, newRes = 2*oldRes - 1 (weights 0/0.5).
// ---------------------------------------------------------------------------
__global__ void upsample_kernel(const float* __restrict__ in, float* __restrict__ out,
                                int oldRes, int newRes)
{
    const int i = blockIdx.x * blockDim.x + threadIdx.x;
    const int total = newRes * newRes;
    if (i >= total) return;
    const int X = i % newRes, Y = i / newRes;
    const int x0 = X >> 1, y0 = Y >> 1;
    const int x1 = min(x0 + 1, oldRes - 1);
    const int y1 = min(y0 + 1, oldRes - 1);
    const float wx = (X & 1) ? 0.5f : 0.0f;
    const float wy = (Y & 1) ? 0.5f : 0.0f;
    const float v00 = in[y0 * oldRes + x0];
    const float v01 = in[y0 * oldRes + x1];
    const float v10 = in[y1 * oldRes + x0];
    const float v11 = in[y1 * oldRes + x1];
    const float top = v00 + wy * (v10 - v00);
    const float bot = v01 + wy * (v11 - v01);
    out[i] = top + wx * (bot - top);
}

// ---------------------------------------------------------------------------
// Top-k selection by histogram radix-select over key = bits(|occ - 0.5|).
// Keys are in [0, 0.5] -> bits < 0x3F000001 -> bin = bits>>17 < 16384.
// ---------------------------------------------------------------------------
__global__ void zero_kernel(int* __restrict__ p, int n)
{
    const int i = blockIdx.x * blockDim.x + threadIdx.x;
    if (i < n) p[i] = 0;
}

__global__ void hist_kernel(const float* __restrict__ occ, int n, int* __restrict__ hist)
{
    const int i = blockIdx.x * blockDim.x + threadIdx.x;
    if (i >= n) return;
    const float u = fabsf(occ[i] - 0.5f);
    const unsigned key = __float_as_uint(u);
    atomicAdd(&hist[key >> 17], 1);
}

// One block, 256 threads; finds first bin T with cum(<=T) >= k and c0 = cum(<T).
__global__ void scan_kernel(const int* __restrict__ hist, int* __restrict__ meta, int k)
{
    __shared__ int sums[256];
    __shared__ int before[256];
    const int t = threadIdx.x;
    int local = 0;
    for (int j = 0; j < NBINS / 256; ++j) local += hist[t * (NBINS / 256) + j];
    sums[t] = local;
    __syncthreads();
    if (t == 0) {
        int run = 0;
        for (int i = 0; i < 256; ++i) { before[i] = run; run += sums[i]; }
        meta[2] = 0;   // below-threshold append counter
        meta[3] = 0;   // boundary-bin counter
    }
    __syncthreads();
    const int b = before[t];
    if (b < k && b + sums[t] >= k) {
        int cum = b;
        for (int j = 0; j < NBINS / 256; ++j) {
            const int bin = t * (NBINS / 256) + j;
            const int h = hist[bin];
            if (cum + h >= k) { meta[0] = bin; meta[1] = cum; break; }
            cum += h;
        }
    }
}

__global__ void collect_kernel(const float* __restrict__ occ, int n,
                               int* __restrict__ meta, int* __restrict__ idxl, int k)
{
    const int i = blockIdx.x * blockDim.x + threadIdx.x;
    if (i >= n) return;
    const int T = meta[0];
    const int c0 = meta[1];
    const float u = fabsf(occ[i] - 0.5f);
    const int bin = (int)(__float_as_uint(u) >> 17);
    if (bin < T) {
        const int pos = atomicAdd(&meta[2], 1);
        idxl[pos] = i;
    } else if (bin == T) {
        const int slot = atomicAdd(&meta[3], 1);
        if (c0 + slot < k) idxl[c0 + slot] = i;
    }
}

// ---------------------------------------------------------------------------
// All 8 batch planes are identical -> replicate plane 0.
// ---------------------------------------------------------------------------
__global__ void replicate_kernel(const float* __restrict__ src, float* __restrict__ dst, int n)
{
    const int i = blockIdx.x * blockDim.x + threadIdx.x;
    if (i >= n) return;
    const float v = src[i];
#pragma unroll
    for (int b = 0; b < 8; ++b) dst[(size_t)b * n + i] = v;
}

// ---------------------------------------------------------------------------

extern "C" void kernel_launch(void* const* d_in, const int* in_sizes, int n_in,
                              void* d_out, int out_size, void* d_ws, size_t ws_size,
                              hipStream_t stream)
{
    const float* w1 = (const float*)d_in[0];   // [2,256]
    const float* b1 = (const float*)d_in[1];   // [256]
    const float* w2 = (const float*)d_in[2];   // [256,1]
    const float* b2 = (const float*)d_in[3];   // [1]
    float* out = (float*)d_out;                // [8,1,1025,1025]

    const size_t need = (size_t)(2 * PLANE) * sizeof(float) +
                        (size_t)(NBINS + 8 + 262144) * sizeof(int);
    if (ws_size < need) return;

    float* occA = (float*)d_ws;
    float* occB = occA + PLANE;
    int* hist = (int*)(occB + PLANE);
    int* meta = hist + NBINS;
    int* idxl = meta + 8;

    // Level 0: evaluate the 65x65 grid (stride 16 in final-pixel space).
    {
        const int n = 65 * 65;
        const int waves = (n + 15) / 16;
        const int blocks = (waves + 7) / 8;
        mlp_eval_kernel<<<blocks, 256, 0, stream>>>(w1, b1, w2, b2, nullptr, n, 65,
                                                    16.0f, occA);
    }

    struct Lvl { int res; int k; float stride; };
    const Lvl lv[4] = { {129, 4096, 8.0f}, {257, 16384, 4.0f},
                        {513, 65536, 2.0f}, {1025, 262144, 1.0f} };

    float* cur = occA;
    float* nxt = occB;
    int oldRes = 65;
    for (int i = 0; i < 4; ++i) {
        const int res = lv[i].res;
        const int n = res * res;
        const int k = lv[i].k;

        upsample_kernel<<<(n + 255) / 256, 256, 0, stream>>>(cur, nxt, oldRes, res);
        zero_kernel<<<(NBINS + 8 + 255) / 256, 256, 0, stream>>>(hist, NBINS + 8);
        hist_kernel<<<(n + 255) / 256, 256, 0, stream>>>(nxt, n, hist);
        scan_kernel<<<1, 256, 0, stream>>>(hist, meta, k);
        collect_kernel<<<(n + 255) / 256, 256, 0, stream>>>(nxt, n, meta, idxl, k);

        const int waves = k / 16;             // k is a multiple of 16
        const int blocks = (waves + 7) / 8;
        mlp_eval_kernel<<<blocks, 256, 0, stream>>>(w1, b1, w2, b2, idxl, k, res,
                                                    lv[i].stride, nxt);

        float* t = cur; cur = nxt; nxt = t;
        oldRes = res;
    }

    // cur == occA holds the final 1025x1025 plane; replicate across batch.
    replicate_kernel<<<(PLANE + 255) / 256, 256, 0, stream>>>(cur, out, PLANE);
}